// TI_MPS_21629455303210
// MI455X (gfx1250) — compile-verified
//
#include <hip/hip_runtime.h>

typedef float v2f __attribute__((ext_vector_type(2)));
typedef float v8f __attribute__((ext_vector_type(8)));

#define BATCH 32
#define SEQ   4096
#define BOND  32
#define FEAT  4
#define NOUT  10

#define NCHUNK 32
#define CHUNK  (SEQ / NCHUNK)      // 128 sites per chunk
#define WPB    4                   // waves per block
#define SPS    34                  // padded LDS row stride (floats), even => 8B-aligned b64 loads

// ---------------------------------------------------------------------------
// Kernel 1: per (batch, chunk) wave computes P = prod_{s in chunk} (I + E_s),
// E_s = sum_i x[b,s,i] * C_i, using V_WMMA_F32_16X16X4_F32 (exact fp32).
// P kept in C/D layout (2x2 tiles of v8f); each step it is staged through LDS
// to regather the A-operand layout; B operand (E) is built from the core
// tensor in LDS (FEAT=4 coefficients contiguous -> one b128 load per element).
// Identity term handled by WMMA accumulation: P <- P*E + P = P*(I+E).
// ---------------------------------------------------------------------------
__global__ __launch_bounds__(32 * WPB) void mps_chunk_kernel(
    const float* __restrict__ input,   // [BATCH, SEQ, FEAT]
    const float* __restrict__ core,    // [BOND, BOND, FEAT]
    float* __restrict__ chunkP)        // [BATCH*NCHUNK, BOND, BOND]
{
    __shared__ __align__(16) float sCore[BOND * BOND * FEAT];   // 16 KB, straight copy
    __shared__ __align__(16) float sP[WPB][BOND * SPS];         // per-wave A-staging

    const int tid = threadIdx.x;
    // cooperative load of core into LDS
    #pragma unroll
    for (int i = 0; i < (BOND * BOND * FEAT) / (32 * WPB); ++i)
        sCore[tid + i * (32 * WPB)] = core[tid + i * (32 * WPB)];
    __syncthreads();

    const int wave = tid >> 5;
    const int lane = tid & 31;
    const int task = blockIdx.x * WPB + wave;     // 0 .. BATCH*NCHUNK-1
    const int b    = task / NCHUNK;
    const int c    = task % NCHUNK;

    const int hl = lane >> 4;    // half-wave (0/1)
    const int ml = lane & 15;

    float* sPw = sP[wave];

    // P = Identity in C/D layout: pc[ti][tj][v] holds (row=16*ti+v+8*hl, col=16*tj+ml)
    v8f pc[2][2];
    #pragma unroll
    for (int ti = 0; ti < 2; ++ti)
        #pragma unroll
        for (int tj = 0; tj < 2; ++tj)
            #pragma unroll
            for (int v = 0; v < 8; ++v) {
                const int row = 16 * ti + v + 8 * hl;
                const int col = 16 * tj + ml;
                pc[ti][tj][v] = (row == col) ? 1.0f : 0.0f;
            }

    const float4* xin = (const float4*)input + (size_t)b * SEQ + (size_t)c * CHUNK;

    for (int s = 0; s < CHUNK; ++s) {
        const float4 xv = xin[s];   // uniform across the wave (broadcast load)

        // (1) stage P (C-layout regs) into LDS row-major with padded stride
        #pragma unroll
        for (int ti = 0; ti < 2; ++ti)
            #pragma unroll
            for (int tj = 0; tj < 2; ++tj)
                #pragma unroll
                for (int v = 0; v < 8; ++v) {
                    const int row = 16 * ti + v + 8 * hl;
                    const int col = 16 * tj + ml;
                    sPw[row * SPS + col] = pc[ti][tj][v];
                }

        // (2) regather P as A-operand slices: a[ti][ks] covers rows 16*ti..+15,
        //     k = 4*ks + 2*hl + {0,1}; lane = m + 16*(k>=2); 8B-aligned b64 loads
        v2f a[2][8];
        #pragma unroll
        for (int ti = 0; ti < 2; ++ti)
            #pragma unroll
            for (int ks = 0; ks < 8; ++ks) {
                const int row = 16 * ti + ml;
                const int k   = 4 * ks + 2 * hl;
                a[ti][ks] = *(const v2f*)&sPw[row * SPS + k];
            }

        // (3) per k-slice: build E's B tiles from core (b128 load gives the 4
        //     FEAT coefficients contiguously), then 4 WMMAs (2x2 output tiles)
        #pragma unroll
        for (int ks = 0; ks < 8; ++ks) {
            v2f bt[2];
            #pragma unroll
            for (int tj = 0; tj < 2; ++tj)
                #pragma unroll
                for (int v = 0; v < 2; ++v) {
                    const int k = 4 * ks + 2 * hl + v;
                    const int n = 16 * tj + ml;
                    const float4 cv = *(const float4*)&sCore[(k * BOND + n) * FEAT];
                    bt[tj][v] = xv.x * cv.x + xv.y * cv.y + xv.z * cv.z + xv.w * cv.w;
                }
            #pragma unroll
            for (int ti = 0; ti < 2; ++ti)
                #pragma unroll
                for (int tj = 0; tj < 2; ++tj)
                    pc[ti][tj] = __builtin_amdgcn_wmma_f32_16x16x4_f32(
                        false, a[ti][ks], false, bt[tj],
                        (short)0, pc[ti][tj], false, false);
        }
    }

    // write chunk product row-major
    float* outP = chunkP + (size_t)task * (BOND * BOND);
    #pragma unroll
    for (int ti = 0; ti < 2; ++ti)
        #pragma unroll
        for (int tj = 0; tj < 2; ++tj)
            #pragma unroll
            for (int v = 0; v < 8; ++v) {
                const int row = 16 * ti + v + 8 * hl;
                const int col = 16 * tj + ml;
                outP[row * BOND + col] = pc[ti][tj][v];
            }
}

// ---------------------------------------------------------------------------
// Kernel 2: per batch, fold the 32 chunk matrices into the vector sequentially
// (32 tiny matvecs), then apply the terminal core. Negligible cost.
// ---------------------------------------------------------------------------
__global__ __launch_bounds__(32) void mps_combine_kernel(
    const float* __restrict__ chunkP,
    const float* __restrict__ init_vec,
    const float* __restrict__ terminal,   // [BOND, NOUT]
    float* __restrict__ out)              // [BATCH, NOUT]
{
    __shared__ float sv[BOND];
    const int lane = threadIdx.x;
    const int b    = blockIdx.x;

    sv[lane] = init_vec[lane];
    __syncthreads();

    for (int c = 0; c < NCHUNK; ++c) {
        const float* P = chunkP + ((size_t)b * NCHUNK + c) * (BOND * BOND);
        float acc = 0.0f;
        #pragma unroll
        for (int l = 0; l < BOND; ++l)
            acc += sv[l] * P[l * BOND + lane];   // coalesced in lane
        __syncthreads();
        sv[lane] = acc;
        __syncthreads();
    }

    if (lane < NOUT) {
        float acc = 0.0f;
        #pragma unroll
        for (int l = 0; l < BOND; ++l)
            acc += sv[l] * terminal[l * NOUT + lane];
        out[b * NOUT + lane] = acc;
    }
}

// ---------------------------------------------------------------------------
extern "C" void kernel_launch(void* const* d_in, const int* in_sizes, int n_in,
                              void* d_out, int out_size, void* d_ws, size_t ws_size,
                              hipStream_t stream) {
    const float* input_data = (const float*)d_in[0];  // [32,4096,4]
    const float* core       = (const float*)d_in[1];  // [32,32,4]
    const float* init_vec   = (const float*)d_in[2];  // [32]
    const float* terminal   = (const float*)d_in[3];  // [32,10]

    float* chunkP = (float*)d_ws;   // needs BATCH*NCHUNK*32*32*4 = 4 MB scratch

    mps_chunk_kernel<<<(BATCH * NCHUNK) / WPB, 32 * WPB, 0, stream>>>(
        input_data, core, chunkP);
    mps_combine_kernel<<<BATCH, 32, 0, stream>>>(
        chunkP, init_vec, terminal, (float*)d_out);
}